// WSLCM_26886495273258
// MI455X (gfx1250) — compile-verified
//
#include <hip/hip_runtime.h>
#include <hip/hip_bf16.h>
#include <math.h>

// ---------------------------------------------------------------------------
// WSLCM on MI455X (gfx1250, wave32).
//
// Box sums (avgpool) are computed as two banded fp32 GEMMs per scale routed
// through V_WMMA_F32_16X16X4_F32 (full fp32 precision on the matrix pipes).
// Rank filter (top-9 mean) is a branchless sorted-insert in VGPRs (min/max
// compare-swap chain, no spills). All data is L2-resident (<5 MB total).
// ---------------------------------------------------------------------------

#define H 384
#define W 384
#define NPIX (H * W)

typedef __attribute__((ext_vector_type(2))) float v2f;
typedef __attribute__((ext_vector_type(8))) float v8f;

__device__ __forceinline__ int refl(int t, int n) {
    // single reflection is enough for |offset| <= 33 << n
    if (t < 0) t = -t;
    if (t >= n) t = 2 * n - 2 - t;
    return t;
}

// ---- prep: inp = x*255, out = -inf ----------------------------------------
__global__ void k_prep(const float* __restrict__ x, float* __restrict__ inp,
                       float* __restrict__ out) {
    int i = blockIdx.x * blockDim.x + threadIdx.x;
    if (i < NPIX) {
        inp[i] = x[i] * 255.0f;
        out[i] = -3.402823466e38f;
    }
}

// ---- gaussian 3x3 (reflect) + 1 -------------------------------------------
__global__ void k_gauss(const float* __restrict__ inp, float* __restrict__ g) {
    int x = blockIdx.x * 16 + threadIdx.x;
    int y = blockIdx.y * 16 + threadIdx.y;
    if (x >= W || y >= H) return;
    int ym = refl(y - 1, H), yp = refl(y + 1, H);
    int xm = refl(x - 1, W), xp = refl(x + 1, W);
    const float* r0 = inp + ym * W;
    const float* r1 = inp + y * W;
    const float* r2 = inp + yp * W;
    float s = r0[xm] + 2.0f * r0[x] + r0[xp]
            + 2.0f * r1[xm] + 4.0f * r1[x] + 2.0f * r1[xp]
            + r2[xm] + 2.0f * r2[x] + r2[xp];
    g[y * W + x] = s * (1.0f / 16.0f) + 1.0f;
}

// ---- build banded reflect box matrix B[k*W + j] ---------------------------
__global__ void k_buildB(float* __restrict__ Bm, int p) {
    int i = blockIdx.x * blockDim.x + threadIdx.x;
    if (i >= NPIX) return;
    int k = i / W, j = i % W;
    int cnt = 0;
    for (int t = j - p; t <= j + p; ++t) cnt += (refl(t, W) == k) ? 1 : 0;
    Bm[i] = (float)cnt;
}

// ---- fp32 WMMA GEMM: C = op(A) * B, all 384x384, one wave per 16x16 tile --
template <bool TRANSA>
__global__ void k_gemm384(const float* __restrict__ A, const float* __restrict__ B,
                          float* __restrict__ C) {
    const int K = 384;
    int lane = threadIdx.x;      // 0..31 (wave32)
    int half = lane >> 4;        // 0 or 1
    int r = lane & 15;
    int tm = blockIdx.x * 16;    // C row base
    int tn = blockIdx.y * 16;    // C col base

    v8f acc = {};
    for (int k0 = 0; k0 < K; k0 += 4) {
        int ka = k0 + 2 * half;  // lanes 0-15: K={k0,k0+1}; lanes 16-31: K={k0+2,k0+3}
        v2f a, b;
        if (TRANSA) {
            a.x = A[ka * K + tm + r];
            a.y = A[(ka + 1) * K + tm + r];
        } else {
            a.x = A[(tm + r) * K + ka];
            a.y = A[(tm + r) * K + ka + 1];
        }
        b.x = B[ka * K + tn + r];
        b.y = B[(ka + 1) * K + tn + r];
        // D = A(16x4) * B(4x16) + C, fp32 in / fp32 accumulate
        acc = __builtin_amdgcn_wmma_f32_16x16x4_f32(
            /*neg_a=*/false, a, /*neg_b=*/false, b,
            /*c_mod=*/(short)0, acc, /*reuse_a=*/false, /*reuse_b=*/false);
    }
    int row = tm + 8 * half;     // lanes 0-15 -> M 0..7, lanes 16-31 -> M 8..15
    int col = tn + r;
#pragma unroll
    for (int v = 0; v < 8; ++v) C[(row + v) * K + col] = acc[v];
}

// ---- top-9 mean rank filter over (2p+1)^2 reflect window -------------------
__global__ void k_rank9(const float* __restrict__ inp, float* __restrict__ M, int p) {
    int x = blockIdx.x * 16 + threadIdx.x;
    int y = blockIdx.y * 16 + threadIdx.y;
    if (x >= W || y >= H) return;
    const float NEG = -3.402823466e38f;
    float t0 = NEG, t1 = NEG, t2 = NEG, t3 = NEG, t4 = NEG,
          t5 = NEG, t6 = NEG, t7 = NEG, t8 = NEG;
    for (int dy = -p; dy <= p; ++dy) {
        const float* row = inp + refl(y + dy, H) * W;
        for (int dx = -p; dx <= p; ++dx) {
            float v = row[refl(x + dx, W)];
            if (v > t0) {
                // replace current min with v, then bubble to restore order
                t0 = v;
                float a;
#define CSWAP(lo, hi) a = fminf(lo, hi); hi = fmaxf(lo, hi); lo = a;
                CSWAP(t0, t1) CSWAP(t1, t2) CSWAP(t2, t3) CSWAP(t3, t4)
                CSWAP(t4, t5) CSWAP(t5, t6) CSWAP(t6, t7) CSWAP(t7, t8)
#undef CSWAP
            }
        }
    }
    M[y * W + x] = (t0 + t1 + t2 + t3 + t4 + t5 + t6 + t7 + t8) * (1.0f / 9.0f);
}

// ---- IRIL = M - Box/(s*s) --------------------------------------------------
__global__ void k_iril(const float* __restrict__ M, const float* __restrict__ Box,
                       float* __restrict__ IRIL, float inv_ss) {
    int i = blockIdx.x * blockDim.x + threadIdx.x;
    if (i < NPIX) IRIL[i] = M[i] - Box[i] * inv_ss;
}

// ---- fused SLCM * W, out = max(out, val) ----------------------------------
__global__ void k_fuse(const float* __restrict__ M, const float* __restrict__ IRIL,
                       const float* __restrict__ g, float* __restrict__ out, int s) {
    int x = blockIdx.x * 16 + threadIdx.x;
    int y = blockIdx.y * 16 + threadIdx.y;
    if (x >= W || y >= H) return;
    int ym = refl(y - s, H), yp = refl(y + s, H);
    int xm = refl(x - s, W), xp = refl(x + s, W);

    // BE: max of 8 non-center neighbors of M at displacement +/- s
    float be = M[ym * W + xm];
    be = fmaxf(be, M[ym * W + x]);
    be = fmaxf(be, M[ym * W + xp]);
    be = fmaxf(be, M[y * W + xm]);
    be = fmaxf(be, M[y * W + xp]);
    be = fmaxf(be, M[yp * W + xm]);
    be = fmaxf(be, M[yp * W + x]);
    be = fmaxf(be, M[yp * W + xp]);

    float gc = g[y * W + x];
    float slcm = fmaxf((be / gc - 1.0f) * gc, 0.0f);

    // IRIL center + 8 neighbors
    float wt = IRIL[y * W + x];
    float n0 = IRIL[ym * W + xm], n1 = IRIL[ym * W + x], n2 = IRIL[ym * W + xp];
    float n3 = IRIL[y * W + xm],  n4 = IRIL[y * W + xp];
    float n5 = IRIL[yp * W + xm], n6 = IRIL[yp * W + x], n7 = IRIL[yp * W + xp];

    float mx = fmaxf(fmaxf(fmaxf(n0, n1), fmaxf(n2, n3)),
                     fmaxf(fmaxf(n4, n5), fmaxf(n6, n7)));
    float sum = n0 + n1 + n2 + n3 + n4 + n5 + n6 + n7;
    float sum2 = n0 * n0 + n1 * n1 + n2 * n2 + n3 * n3 +
                 n4 * n4 + n5 * n5 + n6 * n6 + n7 * n7;

    float wd = fmaxf(wt - mx, 0.0f);
    float mean = sum * (1.0f / 8.0f);
    float var = (sum2 - 8.0f * mean * mean) * (1.0f / 7.0f);  // ddof=1
    float wb = fmaxf(sqrtf(fmaxf(var, 0.0f)), 5.0f);
    float w = wt * wd / wb;
    float val = slcm * w;

    int i = y * W + x;
    out[i] = fmaxf(out[i], val);
}

extern "C" void kernel_launch(void* const* d_in, const int* in_sizes, int n_in,
                              void* d_out, int out_size, void* d_ws, size_t ws_size,
                              hipStream_t stream) {
    const float* x = (const float*)d_in[0];
    float* out = (float*)d_out;
    float* ws = (float*)d_ws;

    float* inp    = ws + 0 * NPIX;
    float* g      = ws + 1 * NPIX;
    float* Bm     = ws + 2 * NPIX;
    float* RowSum = ws + 3 * NPIX;
    float* Box    = ws + 4 * NPIX;
    float* M      = ws + 5 * NPIX;
    float* IRIL   = ws + 6 * NPIX;

    dim3 b1(256), g1((NPIX + 255) / 256);
    dim3 b2(16, 16), g2(W / 16, H / 16);
    dim3 bw(32), gw(H / 16, W / 16);  // one wave per 16x16 C tile

    k_prep<<<g1, b1, 0, stream>>>(x, inp, out);
    k_gauss<<<g2, b2, 0, stream>>>(inp, g);

    const int scales[4] = {3, 5, 23, 33};
    for (int si = 0; si < 4; ++si) {
        int s = scales[si];
        int p = s / 2;
        k_buildB<<<g1, b1, 0, stream>>>(Bm, p);
        // RowSum = inp * B  (horizontal box sum with reflect)
        k_gemm384<false><<<gw, bw, 0, stream>>>(inp, Bm, RowSum);
        // Box = B^T * RowSum (vertical box sum with reflect)
        k_gemm384<true><<<gw, bw, 0, stream>>>(Bm, RowSum, Box);
        // M = mean of top-9 in s x s window (s=3: mean of all 9)
        k_rank9<<<g2, b2, 0, stream>>>(inp, M, p);
        k_iril<<<g1, b1, 0, stream>>>(M, Box, IRIL, 1.0f / (float)(s * s));
        k_fuse<<<g2, b2, 0, stream>>>(M, IRIL, g, out, s);
    }
}